// Net_42288247996587
// MI455X (gfx1250) — compile-verified
//
#include <hip/hip_runtime.h>
#include <hip/hip_bf16.h>

typedef float v2f __attribute__((ext_vector_type(2)));
typedef float v8f __attribute__((ext_vector_type(8)));

#define NN   50000
#define EE   500000
#define EN   550000
#define GG   512
#define HH   5
#define CC   50
#define HC   250
#define HCP  256      // padded H*C
#define FIN  71
#define EDIM 41
#define FPD  161
#define ZD   211      // C + FP
#define ZDP  212      // padded to mult of 4

// ---- monotonic float <-> uint encoding for atomic max ----
__device__ __forceinline__ unsigned fenc(float f) {
    unsigned u = __float_as_uint(f);
    return (u & 0x80000000u) ? ~u : (u | 0x80000000u);
}
__device__ __forceinline__ float fdec(unsigned u) {
    if (u == 0u) return -3.4e38f;
    return (u & 0x80000000u) ? __uint_as_float(u & 0x7FFFFFFFu)
                             : __uint_as_float(~u);
}

// =====================================================================
// Prep: pack row-major B[K][Nc] into WMMA pair layout, zero-padded:
//   dst[p*Ncp + n] = { B[2p][n], B[2p+1][n] },  p < Kp/2, n < Ncp
// =====================================================================
__global__ void pack_b_kernel(const float* __restrict__ src, v2f* __restrict__ dst,
                              int K, int Nc, int Ncp, int Phalf) {
    int i = blockIdx.x * blockDim.x + threadIdx.x;
    if (i >= Phalf * Ncp) return;
    int p = i / Ncp, n = i - p * Ncp;
    int r0 = 2 * p;
    v2f v;
    v.x = (r0     < K && n < Nc) ? src[(size_t)r0       * Nc + n] : 0.f;
    v.y = (r0 + 1 < K && n < Nc) ? src[(size_t)(r0 + 1) * Nc + n] : 0.f;
    dst[i] = v;
}

__global__ void pad_vec_kernel(const float* __restrict__ src, float* __restrict__ dst,
                               int n, int np) {
    int i = blockIdx.x * blockDim.x + threadIdx.x;
    if (i < np) dst[i] = (i < n) ? src[i] : 0.f;
}

// =====================================================================
// K0: column means of edge_attr
// =====================================================================
__global__ void eamean_kernel(const float* __restrict__ ea, float* __restrict__ ea_mean) {
    __shared__ float red[256];
    int col = blockIdx.x;
    float s = 0.f;
    for (int e = threadIdx.x; e < EE; e += blockDim.x)
        s += ea[(size_t)e * EDIM + col];
    red[threadIdx.x] = s;
    __syncthreads();
    for (int off = 128; off > 0; off >>= 1) {
        if (threadIdx.x < off) red[threadIdx.x] += red[threadIdx.x + off];
        __syncthreads();
    }
    if (threadIdx.x == 0) ea_mean[col] = red[0] / (float)EE;
}

// =====================================================================
// Branch-free fp32 WMMA GEMM: C[M,Ncp] = act(A[M,K] @ Bp + bias)
// Requirements: M % 16 == 0, Ncp % 16 == 0, Bp packed/padded, bias padded.
// Inner loop: ds_load_b64 (A) + global_load_b64 (B) + v_wmma per step.
// =====================================================================
__global__ __launch_bounds__(256)
void gemm16_f32(const float* __restrict__ A, int lda,
                const v2f* __restrict__ Bp,
                const float* __restrict__ bias,
                float* __restrict__ C, int ldc,
                int K, int Ncp, int act) {
    __shared__ float sA[16 * 522];          // Kp up to 520, even stride, 8B aligned
    const int m0 = blockIdx.x * 16;
    const int Kp = (K + 3) & ~3;
    for (int idx = threadIdx.x; idx < 16 * Kp; idx += 256) {
        int r = idx / Kp, k = idx - r * Kp;
        sA[r * 522 + k] = (k < K) ? A[(size_t)(m0 + r) * lda + k] : 0.f;
    }
    __syncthreads();
    const int lane  = threadIdx.x & 31, wave = threadIdx.x >> 5;
    const int mrow  = lane & 15;
    const int khalf = lane >> 4;
    const int chi   = khalf * 8;
    const float* sa = &sA[mrow * 522 + khalf * 2];
    const int ntiles = Ncp >> 4;
    for (int nt = wave; nt < ntiles; nt += 8) {
        int n = nt * 16 + (lane & 15);
        float bv = bias ? bias[n] : 0.f;
        v8f c;
#pragma unroll
        for (int i = 0; i < 8; ++i) c[i] = bv;
        const v2f* bp = Bp + (size_t)khalf * Ncp + n;
        for (int k = 0; k < Kp; k += 4) {
            v2f a = *(const v2f*)(sa + k);
            v2f b = bp[(size_t)(k >> 1) * Ncp];
            c = __builtin_amdgcn_wmma_f32_16x16x4_f32(false, a, false, b,
                                                      (short)0, c, false, false);
        }
#pragma unroll
        for (int i = 0; i < 8; ++i) {
            float v = c[i];
            if (act) v = fmaxf(v, 0.f);
            C[(size_t)(m0 + chi + i) * ldc + n] = v;
        }
    }
}

// =====================================================================
// Fused edge kernel: 16 edges/block.  ee = ea@We (WMMA, packed We);
// logits[e][h] = sum_c att * lrelu(ee + xl[src] + xr[dst])  (ds_add_f32)
// xl/xr row gathers go straight to LDS via global_load_async_to_lds_b128
// (ASYNCcnt-tracked, VGPR-bypass), synced with s_wait_asynccnt + barrier.
// EN % 16 == 0 -> no tail guards.
// =====================================================================
__global__ __launch_bounds__(256)
void edge_logits_kernel(const int* __restrict__ src, const int* __restrict__ dst,
                        const float* __restrict__ edge_attr,
                        const float* __restrict__ ea_mean,
                        const float* __restrict__ xlp, const float* __restrict__ xrp,
                        const v2f* __restrict__ WeP, const float* __restrict__ attP,
                        float* __restrict__ logits) {
    __shared__ __align__(16) float s_xl[16][HCP];
    __shared__ __align__(16) float s_xr[16][HCP];
    __shared__ float s_ea[16][46];          // Kp=44, even stride 46
    __shared__ int   s_src[16], s_dst[16];
    __shared__ float s_lg[16][8];
    const int tid = threadIdx.x;
    const int te  = blockIdx.x * 16;
    if (tid < 16) {
        int e = te + tid;
        int s, d;
        if (e < EE) { s = src[e]; d = dst[e]; } else { s = e - EE; d = e - EE; }
        s_src[tid] = s; s_dst[tid] = d;
    }
    if (tid < 128) ((float*)s_lg)[tid] = 0.f;
    __syncthreads();
    // ---- async gather of xl[src], xr[dst] rows into LDS (b128 chunks) ----
    {
        unsigned lb_xl = (unsigned)(uintptr_t)&s_xl[0][0];
        unsigned lb_xr = (unsigned)(uintptr_t)&s_xr[0][0];
#pragma unroll
        for (int it = 0; it < 4; ++it) {
            int v  = tid + it * 256;            // 16 rows * 64 b128-chunks
            int m  = v >> 6;
            int c4 = (v & 63) << 2;
            unsigned loff = (unsigned)((m * HCP + c4) * 4);
            const float* ga = xlp + (size_t)s_src[m] * HCP + c4;
            const float* gb = xrp + (size_t)s_dst[m] * HCP + c4;
            asm volatile("global_load_async_to_lds_b128 %0, %1, off"
                         :: "v"(lb_xl + loff),
                            "v"((unsigned long long)(uintptr_t)ga) : "memory");
            asm volatile("global_load_async_to_lds_b128 %0, %1, off"
                         :: "v"(lb_xr + loff),
                            "v"((unsigned long long)(uintptr_t)gb) : "memory");
        }
    }
    // ---- edge-attr tile via normal path (small) ----
    for (int idx = tid; idx < 16 * 44; idx += 256) {
        int m = idx / 44, k = idx - m * 44;
        int e = te + m;
        float v = 0.f;
        if (k < EDIM) v = (e < EE) ? edge_attr[(size_t)e * EDIM + k] : ea_mean[k];
        s_ea[m][k] = v;
    }
    asm volatile("s_wait_asynccnt 0" ::: "memory");
    __syncthreads();
    const int lane  = tid & 31, wave = tid >> 5;
    const int mrow  = lane & 15;
    const int khalf = lane >> 4;
    const int chi   = khalf * 8;
    const float* sa = &s_ea[mrow][khalf * 2];
    for (int nt = wave; nt < 16; nt += 8) {
        int n = nt * 16 + (lane & 15);
        v8f c;
#pragma unroll
        for (int i = 0; i < 8; ++i) c[i] = 0.f;
        const v2f* bp = WeP + (size_t)khalf * HCP + n;
#pragma unroll
        for (int k = 0; k < 44; k += 4) {
            v2f a = *(const v2f*)(sa + k);
            v2f b = bp[(k >> 1) * HCP];
            c = __builtin_amdgcn_wmma_f32_16x16x4_f32(false, a, false, b,
                                                      (short)0, c, false, false);
        }
        float av = attP[n];                  // 0 in padded cols -> no-op adds
        int h = n / CC;
        if (h > 7) h = 7;
#pragma unroll
        for (int i = 0; i < 8; ++i) {
            int m = chi + i;
            float v = c[i] + s_xl[m][n] + s_xr[m][n];
            v = (v > 0.f) ? v : 0.2f * v;    // leaky_relu before att dot
            atomicAdd(&s_lg[m][h], v * av);  // ds_add_f32
        }
    }
    __syncthreads();
    if (tid < 16 * HH) {
        int m = tid / HH, h = tid - m * HH;
        logits[(size_t)(te + m) * HH + h] = s_lg[m][h];
    }
}

// =====================================================================
__global__ void segmax_kernel(const float* __restrict__ logits,
                              const int* __restrict__ dst,
                              unsigned* __restrict__ menc) {
    int i = blockIdx.x * blockDim.x + threadIdx.x;
    if (i >= EN * HH) return;
    int e = i / HH, h = i - e * HH;
    int d = (e < EE) ? dst[e] : e - EE;
    atomicMax(&menc[d * HH + h], fenc(logits[i]));
}

__global__ void expv_kernel(float* __restrict__ logits,
                            const int* __restrict__ dst,
                            const unsigned* __restrict__ menc,
                            float* __restrict__ denom) {
    int i = blockIdx.x * blockDim.x + threadIdx.x;
    if (i >= EN * HH) return;
    int e = i / HH, h = i - e * HH;
    int d = (e < EE) ? dst[e] : e - EE;
    float ex = expf(logits[i] - fdec(menc[d * HH + h]));
    logits[i] = ex;
    atomicAdd(&denom[d * HH + h], ex);
}

__global__ void scatter_kernel(const float* __restrict__ expv,
                               const float* __restrict__ denom,
                               const float* __restrict__ xlp,
                               const int* __restrict__ src,
                               const int* __restrict__ dst,
                               float* __restrict__ outacc) {
    int i = blockIdx.x * blockDim.x + threadIdx.x;
    if (i >= EN * HC) return;
    int e = i / HC, col = i - e * HC;
    int h = col / CC;
    int s, d;
    if (e < EE) { s = src[e]; d = dst[e]; } else { s = e - EE; d = e - EE; }
    float alpha = expv[(size_t)e * HH + h] / (denom[(size_t)d * HH + h] + 1e-16f);
    atomicAdd(&outacc[(size_t)d * HC + col], xlp[(size_t)s * HCP + col] * alpha);
}

__global__ void headmean_pool_kernel(const float* __restrict__ outacc,
                                     const float* __restrict__ bias_o,
                                     const int* __restrict__ batch,
                                     unsigned* __restrict__ pooled_enc) {
    int i = blockIdx.x * blockDim.x + threadIdx.x;
    if (i >= NN * CC) return;
    int n = i / CC, c = i - n * CC;
    float s = 0.f;
#pragma unroll
    for (int h = 0; h < HH; ++h) s += outacc[(size_t)n * HC + h * CC + c];
    float v = fmaxf(s * 0.2f + bias_o[c], 0.f);
    atomicMax(&pooled_enc[batch[n] * CC + c], fenc(v));
}

__global__ void buildz_kernel(const unsigned* __restrict__ pooled_enc,
                              const float* __restrict__ fp,
                              float* __restrict__ z) {
    int i = blockIdx.x * blockDim.x + threadIdx.x;
    if (i >= GG * ZDP) return;
    int g = i / ZDP, c = i - g * ZDP;
    float v = 0.f;
    if (c < CC)       v = fdec(pooled_enc[g * CC + c]);
    else if (c < ZD)  v = fp[g * FPD + (c - CC)];
    z[i] = v;
}

__global__ void final_kernel(const float* __restrict__ z2,
                             const float* __restrict__ W3,
                             const float* __restrict__ b3,
                             float* __restrict__ out) {
    int g = blockIdx.x * blockDim.x + threadIdx.x;
    if (g >= GG) return;
    float s = b3[0];
#pragma unroll
    for (int k = 0; k < 64; ++k) s += z2[g * 64 + k] * W3[k];
    out[g] = s;
}

// =====================================================================
extern "C" void kernel_launch(void* const* d_in, const int* in_sizes, int n_in,
                              void* d_out, int out_size, void* d_ws, size_t ws_size,
                              hipStream_t stream) {
    const float* x      = (const float*)d_in[0];
    const int*   src    = (const int*)d_in[1];
    const int*   dst    = src + EE;
    const float* ea     = (const float*)d_in[2];
    const int*   batch  = (const int*)d_in[3];
    const float* fp     = (const float*)d_in[4];
    const float* Wl     = (const float*)d_in[5];
    const float* bl     = (const float*)d_in[6];
    const float* Wr     = (const float*)d_in[7];
    const float* br     = (const float*)d_in[8];
    const float* We     = (const float*)d_in[9];
    const float* att    = (const float*)d_in[10];
    const float* bias_o = (const float*)d_in[11];
    const float* W1     = (const float*)d_in[12];
    const float* b1     = (const float*)d_in[13];
    const float* W2     = (const float*)d_in[14];
    const float* b2     = (const float*)d_in[15];
    const float* W3     = (const float*)d_in[16];
    const float* b3     = (const float*)d_in[17];

    char* base = (char*)d_ws;
    size_t off = 0;
    auto alloc = [&](size_t bytes) {
        size_t r = off; off += (bytes + 255) & ~(size_t)255; return r;
    };
    float*    xlp    = (float*)   (base + alloc((size_t)NN * HCP * 4));
    float*    xrp    = (float*)   (base + alloc((size_t)NN * HCP * 4));
    float*    logits = (float*)   (base + alloc((size_t)EN * HH * 4));
    unsigned* menc   = (unsigned*)(base + alloc((size_t)NN * HH * 4));
    float*    denom  = (float*)   (base + alloc((size_t)NN * HH * 4));
    float*    outacc = (float*)   (base + alloc((size_t)NN * HC * 4));
    unsigned* pooled = (unsigned*)(base + alloc((size_t)GG * CC * 4));
    float*    z      = (float*)   (base + alloc((size_t)GG * ZDP * 4));
    float*    z1     = (float*)   (base + alloc((size_t)GG * 512 * 4));
    float*    z2     = (float*)   (base + alloc((size_t)GG * 64 * 4));
    float*    eamean = (float*)   (base + alloc((size_t)EDIM * 4));
    // packed/padded weights
    v2f* WlP = (v2f*)(base + alloc((size_t)36  * HCP * 8));   // Kp=72
    v2f* WrP = (v2f*)(base + alloc((size_t)36  * HCP * 8));
    v2f* WeP = (v2f*)(base + alloc((size_t)22  * HCP * 8));   // Kp=44
    v2f* W1P = (v2f*)(base + alloc((size_t)106 * 512 * 8));   // Kp=212
    v2f* W2P = (v2f*)(base + alloc((size_t)256 * 64  * 8));   // Kp=512
    float* blP  = (float*)(base + alloc(HCP * 4));
    float* brP  = (float*)(base + alloc(HCP * 4));
    float* attP = (float*)(base + alloc(HCP * 4));

    hipMemsetAsync(menc,   0, (size_t)NN * HH * 4, stream);
    hipMemsetAsync(denom,  0, (size_t)NN * HH * 4, stream);
    hipMemsetAsync(outacc, 0, (size_t)NN * HC * 4, stream);
    hipMemsetAsync(pooled, 0, (size_t)GG * CC * 4, stream);

    // ---- prep: pack weights / pad vectors ----
    pack_b_kernel<<<(36  * HCP + 255) / 256, 256, 0, stream>>>(Wl, WlP, FIN, HC, HCP, 36);
    pack_b_kernel<<<(36  * HCP + 255) / 256, 256, 0, stream>>>(Wr, WrP, FIN, HC, HCP, 36);
    pack_b_kernel<<<(22  * HCP + 255) / 256, 256, 0, stream>>>(We, WeP, EDIM, HC, HCP, 22);
    pack_b_kernel<<<(106 * 512 + 255) / 256, 256, 0, stream>>>(W1, W1P, ZD, 512, 512, 106);
    pack_b_kernel<<<(256 * 64  + 255) / 256, 256, 0, stream>>>(W2, W2P, 512, 64, 64, 256);
    pad_vec_kernel<<<1, HCP, 0, stream>>>(bl,  blP,  HC, HCP);
    pad_vec_kernel<<<1, HCP, 0, stream>>>(br,  brP,  HC, HCP);
    pad_vec_kernel<<<1, HCP, 0, stream>>>(att, attP, HC, HCP);

    eamean_kernel<<<EDIM, 256, 0, stream>>>(ea, eamean);

    // node transforms (padded outputs, stride 256)
    gemm16_f32<<<NN / 16, 256, 0, stream>>>(x, FIN, WlP, blP, xlp, HCP, FIN, HCP, 0);
    gemm16_f32<<<NN / 16, 256, 0, stream>>>(x, FIN, WrP, brP, xrp, HCP, FIN, HCP, 0);

    edge_logits_kernel<<<EN / 16, 256, 0, stream>>>(src, dst, ea, eamean,
                                                    xlp, xrp, WeP, attP, logits);

    segmax_kernel<<<(EN * HH + 255) / 256, 256, 0, stream>>>(logits, dst, menc);
    expv_kernel<<<(EN * HH + 255) / 256, 256, 0, stream>>>(logits, dst, menc, denom);
    scatter_kernel<<<(EN * HC + 255) / 256, 256, 0, stream>>>(logits, denom, xlp,
                                                              src, dst, outacc);
    headmean_pool_kernel<<<(NN * CC + 255) / 256, 256, 0, stream>>>(
        outacc, bias_o, batch, pooled);
    buildz_kernel<<<(GG * ZDP + 255) / 256, 256, 0, stream>>>(pooled, fp, z);

    // MLP
    gemm16_f32<<<GG / 16, 256, 0, stream>>>(z,  ZDP, W1P, b1, z1, 512, ZDP, 512, 1);
    gemm16_f32<<<GG / 16, 256, 0, stream>>>(z1, 512, W2P, b2, z2, 64,  512, 64,  1);
    final_kernel<<<(GG + 255) / 256, 256, 0, stream>>>(z2, W3, b3, (float*)d_out);
}